// SO3_Linear_11948599018117
// MI455X (gfx1250) — compile-verified
//
#include <hip/hip_runtime.h>

// SO(3) linear layer: out[b,m,o] = sum_i in[b,m,i] * W[l(m),o,i]  (+ bias on m==0)
// B=20000, M=49, IN=OUT=128, 7 distinct weight matrices (per degree l).
//
// Memory-bound problem (~1.0 GB moved, 32.1 GFLOP -> 43 us floor at 23.3 TB/s).
// Compute path uses V_WMMA_F32_16X16X4_F32 (exact fp32, matrix pipe).
// One workgroup = 8 waves = one 16-row b-tile for one m; wave w owns o-tile w.
// 32 chained WMMAs per wave accumulate K=128.

typedef float v2f __attribute__((ext_vector_type(2)));
typedef float v8f __attribute__((ext_vector_type(8)));

namespace {
constexpr int kB   = 20000;
constexpr int kM   = 49;     // (LMAX+1)^2
constexpr int kIN  = 128;
constexpr int kOUT = 128;
}

__global__ __launch_bounds__(256, 4)
void so3_linear_wmma_kernel(const float* __restrict__ in,     // [B, M, IN]
                            const float* __restrict__ weight, // [7, OUT, IN]
                            const float* __restrict__ bias,   // [OUT]
                            const int*   __restrict__ expand_index, // [M]
                            float* __restrict__ out)          // [B, M, OUT]
{
    const int tid  = threadIdx.x;
    const int lane = tid & 31;          // wave32
    const int wave = tid >> 5;          // 0..7 -> o-tile index
    const int half = lane >> 4;         // 0 or 1 (K sub-pair / M sub-half)
    const int r    = lane & 15;         // row within 16-wide tile

    const int b0 = blockIdx.x * 16;     // b-tile base
    const int m  = blockIdx.y;          // which m row (0..48)
    const int l  = expand_index[m];     // degree -> weight matrix
    const int o0 = wave * 16;           // o-tile base

    // A fragment rows: input[b0+r, m, :], lane half selects K pair {0,1} vs {2,3}
    const float* __restrict__ Arow =
        in + (size_t)(b0 + r) * (kM * kIN) + (size_t)m * kIN + 2 * half;
    // B fragment: B[k][n] = W[l, o0+n, k]; contiguous float2 along i.
    const float* __restrict__ Brow =
        weight + (size_t)l * (kOUT * kIN) + (size_t)(o0 + r) * kIN + 2 * half;

    v8f c = {};  // 16x16 f32 accumulator tile (8 VGPRs)

#pragma unroll
    for (int t = 0; t < kIN / 4; ++t) {   // 32 k-steps of 4
        v2f a = *(const v2f*)(Arow + 4 * t);
        v2f b = *(const v2f*)(Brow + 4 * t);
        // 8 args: (neg_a, A, neg_b, B, c_mod, C, reuse_a, reuse_b)
        c = __builtin_amdgcn_wmma_f32_16x16x4_f32(
                false, a, false, b, (short)0, c, false, false);
    }

    // Bias applies only to the l=0 (m==0) row of the M dimension.
    if (m == 0) {
        const float bv = bias[o0 + r];
#pragma unroll
        for (int v = 0; v < 8; ++v) c[v] += bv;
    }

    // D layout: c[v] -> out row (b0 + v + 8*half), col (o0 + r).
    float* __restrict__ Obase = out + (size_t)m * kOUT + (size_t)(o0 + r);
#pragma unroll
    for (int v = 0; v < 8; ++v) {
        const int row = b0 + v + 8 * half;
        Obase[(size_t)row * (kM * kOUT)] = c[v];
    }
}

extern "C" void kernel_launch(void* const* d_in, const int* in_sizes, int n_in,
                              void* d_out, int out_size, void* d_ws, size_t ws_size,
                              hipStream_t stream)
{
    (void)in_sizes; (void)n_in; (void)out_size; (void)d_ws; (void)ws_size;

    const float* in_emb  = (const float*)d_in[0];  // [B, M, IN]
    const float* weight  = (const float*)d_in[1];  // [7, OUT, IN]
    const float* bias    = (const float*)d_in[2];  // [OUT]
    const int*   eidx    = (const int*)  d_in[3];  // [M]
    float*       out     = (float*)d_out;          // [B, M, OUT]

    dim3 grid(kB / 16, kM, 1);   // 1250 x 49 workgroups
    dim3 block(256, 1, 1);       // 8 waves; wave w -> o-tile w

    so3_linear_wmma_kernel<<<grid, block, 0, stream>>>(in_emb, weight, bias, eidx, out);
}